// BaseGCN_31920196944505
// MI455X (gfx1250) — compile-verified
//
#include <hip/hip_runtime.h>
#include <stdint.h>

typedef __attribute__((ext_vector_type(2))) float v2f;
typedef __attribute__((ext_vector_type(4))) float v4f;
typedef __attribute__((ext_vector_type(8))) float v8f;

#define N_NODES 4096
#define CHAN    3
#define KMAX    20          // reference uses k = 20
#define TILE    16
#define WAVES   16
#define T1      (WAVES * 32)

// ---------------------------------------------------------------------------
// Kernel 1: fused pairwise-distance (WMMA f32 16x16x4, C=3 padded to K=4)
// + per-row top-k selection. One workgroup = 16 rows; 16 waves.
//   LDS: sq[4096] (16KB) + d2[16][4096] (256KB) = 272KB (<= 320KB/WGP)
//
// NOTE: we select on *squared* distances. sqrt is monotone on the clamped
// d2, so the top-k index set (and its order, via the packed-key index
// tie-break) is identical to selecting on sqrt(d2) — and we avoid the
// ~20-VALU-op IEEE sqrt expansion per element that dominated the prior
// version's inner loop.
// ---------------------------------------------------------------------------
__global__ __launch_bounds__(T1) void knn_kernel(const float* __restrict__ x,
                                                 const int*   __restrict__ kptr,
                                                 int*         __restrict__ idx_out) {
  extern __shared__ char smem_raw[];
  float* sqLDS = (float*)smem_raw;                       // 4096 floats
  float* d2LDS = (float*)(smem_raw + N_NODES * 4);       // 16*4096 floats

  const int tid  = threadIdx.x;
  const int lane = tid & 31;
  const int wave = tid >> 5;
  const int b    = blockIdx.x >> 8;                 // 256 row-tiles per batch
  const int i0   = (blockIdx.x & 255) * TILE;
  const float* xb = x + (size_t)b * N_NODES * CHAN;

  // ---- phase 0: squared norms for all 4096 nodes of this batch ----
  for (int n = tid; n < N_NODES; n += T1) {
    float c0 = xb[n * 3 + 0], c1 = xb[n * 3 + 1], c2 = xb[n * 3 + 2];
    sqLDS[n] = c0 * c0 + c1 * c1 + c2 * c2;
  }
  __syncthreads();

  // ---- phase 1: Gram tile via V_WMMA_F32_16X16X4_F32, d2 -> LDS ----
  // A (16x4): lanes 0-15 hold K=0/1 in VGPR0/1, lanes 16-31 hold K=2/3; M=lane%16
  // B (4x16): lanes 0-15 hold K=0/1 rows, lanes 16-31 hold K=2/3 rows; N=lane%16
  {
    const int m = i0 + (lane & 15);
    float ac0 = xb[m * 3 + 0], ac1 = xb[m * 3 + 1], ac2 = xb[m * 3 + 2];
    v2f A;
    A[0] = (lane < 16) ? ac0 : ac2;
    A[1] = (lane < 16) ? ac1 : 0.0f;          // K=3 pad

    for (int t = wave; t < N_NODES / TILE; t += WAVES) {
      const int j0 = t * TILE;
      const int n  = j0 + (lane & 15);
      float bc0 = xb[n * 3 + 0], bc1 = xb[n * 3 + 1], bc2 = xb[n * 3 + 2];
      v2f Bm;
      Bm[0] = (lane < 16) ? bc0 : bc2;
      Bm[1] = (lane < 16) ? bc1 : 0.0f;

      v8f acc = {};
      acc = __builtin_amdgcn_wmma_f32_16x16x4_f32(false, A, false, Bm,
                                                  (short)0, acc, false, false);
      // C/D layout: lane L, VGPR v -> M = v + 8*(L>=16), N = L%16
      const int mbase = (lane >= 16) ? 8 : 0;
      const int j     = j0 + (lane & 15);
      const float sqj = sqLDS[j];
#pragma unroll
      for (int v = 0; v < 8; ++v) {
        const int mm = v + mbase;
        float d2 = sqLDS[i0 + mm] + sqj - 2.0f * acc[v];
        d2LDS[mm * N_NODES + j] = fmaxf(d2, 0.0f);
      }
    }
  }
  __syncthreads();

  // ---- phase 2: top-k per row; wave w handles row w ----
  {
    int kk = *kptr;
    if (kk > KMAX) kk = KMAX;
    if (kk < 1)  kk = 1;

    // packed key: (d2_bits << 32) | col  (d2>=0 -> bits are monotone;
    // low bits = index give jax top_k tie-break: smaller index wins)
    unsigned long long keys[KMAX];
#pragma unroll
    for (int s = 0; s < KMAX; ++s) keys[s] = ~0ULL;

    const float* drow = d2LDS + wave * N_NODES;
    for (int it = 0; it < N_NODES / 128; ++it) {
      const int jb = it * 128 + lane * 4;
      v4f dv = *(const v4f*)(drow + jb);       // ds_load_b128, conflict-free
#pragma unroll
      for (int c = 0; c < 4; ++c) {
        unsigned long long key =
            ((unsigned long long)__float_as_uint(dv[c]) << 32) |
            (unsigned int)(jb + c);
        if (key < keys[KMAX - 1]) {
          keys[KMAX - 1] = key;                // insert + bubble (all constant idx)
#pragma unroll
          for (int s = KMAX - 1; s > 0; --s) {
            unsigned long long lo = keys[s - 1] < keys[s] ? keys[s - 1] : keys[s];
            unsigned long long hi = keys[s - 1] < keys[s] ? keys[s] : keys[s - 1];
            keys[s - 1] = lo;
            keys[s]     = hi;
          }
        }
      }
    }

    const int ig   = i0 + wave;
    int*      orow = idx_out + ((size_t)(b * N_NODES + ig)) * 32;
#pragma unroll
    for (int r = 0; r < KMAX; ++r) {
      // wave-wide min of current heads
      unsigned long long mk = keys[0];
#pragma unroll
      for (int off = 16; off > 0; off >>= 1) {
        unsigned int hi = (unsigned int)(mk >> 32);
        unsigned int lo = (unsigned int)mk;
        hi = __shfl_xor(hi, off, 32);
        lo = __shfl_xor(lo, off, 32);
        unsigned long long o = ((unsigned long long)hi << 32) | lo;
        mk = (o < mk) ? o : mk;
      }
      if (keys[0] == mk) {                     // winner pops its head
#pragma unroll
        for (int s = 0; s < KMAX - 1; ++s) keys[s] = keys[s + 1];
        keys[KMAX - 1] = ~0ULL;
      }
      if (lane == r && r < kk)
        orow[r] = (int)(unsigned int)(mk & 0xffffffffu);
    }
  }
}

// ---------------------------------------------------------------------------
// Kernel 2: single streaming pass over the 256MB output.
//   lap = I - adj/k   (deg == k for every node: top_k always returns k
//   distinct indices incl. self, so dinv_i*dinv_j == 1/k everywhere).
// Each thread emits one float4 with a nontemporal b128 store (output >
// 192MB L2, so NT keeps the write stream from thrashing the cache).
// Block = 256 threads = 1024 columns of one row; 4 blocks per row.
// ---------------------------------------------------------------------------
__global__ __launch_bounds__(256) void lap_kernel(const int* __restrict__ idx_ws,
                                                  const int* __restrict__ kptr,
                                                  float*     __restrict__ out) {
  __shared__ int nb[KMAX];
  const int t    = threadIdx.x;
  const int bid  = blockIdx.x;
  const int rowg = bid >> 2;                 // global row over B*N
  const int seg  = bid & 3;
  const int i    = rowg & (N_NODES - 1);

  int kk = *kptr;
  if (kk > KMAX) kk = KMAX;
  if (kk < 1)  kk = 1;
  if (t < kk) nb[t] = idx_ws[(size_t)rowg * 32 + t];
  __syncthreads();

  const float w  = 1.0f / (float)kk;         // dinv_i * dinv_j
  const int   cq = seg * 256 + t;
  const int   c0 = cq * 4;

  float r0 = (i == c0 + 0) ? 1.0f : 0.0f;
  float r1 = (i == c0 + 1) ? 1.0f : 0.0f;
  float r2 = (i == c0 + 2) ? 1.0f : 0.0f;
  float r3 = (i == c0 + 3) ? 1.0f : 0.0f;

  for (int n = 0; n < kk; ++n) {
    const int j = nb[n];                     // self is in the list -> diag 1-1/k
    if      (j == c0 + 0) r0 -= w;
    else if (j == c0 + 1) r1 -= w;
    else if (j == c0 + 2) r2 -= w;
    else if (j == c0 + 3) r3 -= w;
  }

  v4f v = {r0, r1, r2, r3};
  __builtin_nontemporal_store(v, (v4f*)(out + (size_t)rowg * N_NODES + c0));
}

// ---------------------------------------------------------------------------
extern "C" void kernel_launch(void* const* d_in, const int* in_sizes, int n_in,
                              void* d_out, int out_size, void* d_ws, size_t ws_size,
                              hipStream_t stream) {
  const float* x    = (const float*)d_in[0];   // (B, 4096, 3) float32
  const int*   kptr = (const int*)d_in[1];     // scalar k (device)
  float*       out  = (float*)d_out;           // (B, 4096, 4096) float32
  int*         idxw = (int*)d_ws;              // B*4096*32 ints of scratch

  const int B = in_sizes[0] / (N_NODES * CHAN);

  const size_t smem = (size_t)N_NODES * 4 + (size_t)TILE * N_NODES * 4; // 272 KB
  knn_kernel<<<dim3(B * (N_NODES / TILE)), dim3(T1), smem, stream>>>(x, kptr, idxw);
  lap_kernel<<<dim3(B * N_NODES * 4), dim3(256), 0, stream>>>(idxw, kptr, out);
}